// DeltaModulationEncoder_54099408060857
// MI455X (gfx1250) — compile-verified
//
#include <hip/hip_runtime.h>
#include <stdint.h>

// Delta-modulation encoder for MI455X (gfx1250).
// 2048 independent sequential chains (one per lane); wall-clock bound by the
// per-chain serial recurrence, so the inner step is restructured to a 4-deep
// dependent-VALU chain (bit-identical to the reference) with off-path ops
// filling VOPD dual-issue slots.
// CDNA5 path: triple-buffered global->LDS async copies (ASYNCcnt) turning the
// 64KB-strided per-chain streams into coalesced 256B global bursts, with
// s_wait_asynccnt pipelining two stages of prefetch ahead of the compute.

#define B_   32
#define C_   64
#define T_   16384
#define NCH  (B_ * C_)            // 2048 chains
#define TPB  64                   // threads (=chains) per block, 2 wave32s
#define NBLK (NCH / TPB)          // 32 blocks
#define TT   64                   // time steps staged per chain per stage
#define PITCH 68                  // LDS row pitch in floats (272B: 16B aligned, bank-skewed)
#define NSTAGES (T_ / TT)         // 256
#define CHUNKS_PER_ROW   (TT * 4 / 16)   // 16 x 16B per row
#define CHUNKS_PER_THREAD 16             // async b128 per thread per stage

// One 16B async copy: copies 16 bytes from global address gbase plus g_byte_off to LDS byte offset lds_byte_off.  GVS mode:
// SGPR64 base + per-lane 32-bit byte offset.  Tracked by ASYNCcnt.
__device__ __forceinline__ void async_copy16(uint32_t lds_byte_off,
                                             const void* gbase,
                                             uint32_t g_byte_off) {
  asm volatile("global_load_async_to_lds_b128 %0, %1, %2"
               :: "v"(lds_byte_off), "v"(g_byte_off), "s"(gbase)
               : "memory");
}

// Issue one stage's tile: rows = this block's 64 chains, cols = TT floats.
// chunk = it*TPB + tid; 16 consecutive lanes cover 256 contiguous bytes of a row.
__device__ __forceinline__ void issue_stage(const float* __restrict__ x,
                                            uint32_t lds_base_off,
                                            int block_chain0, int t0, int tid) {
#pragma unroll
  for (int it = 0; it < CHUNKS_PER_THREAD; ++it) {
    const int chunk = it * TPB + tid;
    const int row   = chunk >> 4;   // /CHUNKS_PER_ROW
    const int c16   = chunk & 15;
    const uint32_t goff =
        (uint32_t)(((uint32_t)(block_chain0 + row)) * (uint32_t)(T_ * 4) +
                   (uint32_t)(t0 * 4) + (uint32_t)(c16 * 16));
    const uint32_t loff = lds_base_off + (uint32_t)(row * PITCH * 4 + c16 * 16);
    async_copy16(loff, x, goff);
  }
}

// One recurrence step.  Bit-identical to the reference:
//   e = x - recon; c1 = e > th; c2 = e < -th;
//   recon' = recon + (c1 - c2)*th  ==  c1 ? recon+th : (c2 ? recon-th : recon)
// (net*th is exact for net in {-1,0,1}, so the selected rp/rn equal the
// reference's recon + net*th exactly).  Loop-carried chain: e -> cmp -> cnd -> cnd.
__device__ __forceinline__ float dm_step(float xv, float th, float nth, float& recon) {
  const float e  = xv - recon;        // dep 1
  const float rp = recon + th;        // dep 1 (parallel)
  const float rn = recon - th;        // dep 1 (parallel)
  const bool  c1 = e > th;            // dep 2
  const bool  c2 = e < nth;           // dep 2 (parallel)
  const float t  = c2 ? rn : recon;   // dep 3
  recon          = c1 ? rp : t;       // dep 4
  const float n  = c2 ? -1.0f : 0.0f; // off-path
  return c1 ? 1.0f : n;               // off-path
}

__global__ __launch_bounds__(TPB)
void delta_mod_encoder_kernel(const float* __restrict__ x,
                              const float* __restrict__ thr,
                              float* __restrict__ out) {
  __shared__ __align__(16) float sbuf[3][TPB * PITCH];   // 3 x ~17KB = 52KB

  const int tid    = threadIdx.x;
  const int chain0 = blockIdx.x * TPB;
  const int chain  = chain0 + tid;

  float th = thr[0];
  th = fminf(fmaxf(th, 0.01f), 0.5f);
  const float nth = -th;

  // Workgroup-relative LDS byte offsets (low 32 bits of generic shared ptr).
  const uint32_t l0 = (uint32_t)(uintptr_t)(&sbuf[0][0]);
  const uint32_t l1 = (uint32_t)(uintptr_t)(&sbuf[1][0]);
  const uint32_t l2 = (uint32_t)(uintptr_t)(&sbuf[2][0]);

  float recon = 0.0f;
  float* __restrict__ orow = out + (size_t)chain * T_;

  // Prologue: prefetch stages 0 and 1 two-deep.
  issue_stage(x, l0, chain0, 0 * TT, tid);
  issue_stage(x, l1, chain0, 1 * TT, tid);

  // Rotating buffer offsets: a = stage s, b = s+1 (in flight), c = issue target (s+2).
  uint32_t a = l0, b = l1, c = l2;

  for (int s = 0; s < NSTAGES; ++s) {
    if (s + 2 < NSTAGES)
      issue_stage(x, c, chain0, (s + 2) * TT, tid);

    // Loads complete in order per wave: release stage s while up to two
    // stages (2 x 16 instructions) remain outstanding.
    const int ahead = NSTAGES - 1 - s;
    if (ahead >= 2)      asm volatile("s_wait_asynccnt 32" ::: "memory");
    else if (ahead == 1) asm volatile("s_wait_asynccnt 16" ::: "memory");
    else                 asm volatile("s_wait_asynccnt 0"  ::: "memory");
    __syncthreads();  // publish LDS writes to both waves

    // Compute stage s from buffer `a`.
    {
      const float* __restrict__ rowp =
          (const float*)((const char*)&sbuf[0][0] + (a - l0)) + tid * PITCH;
      float* __restrict__ ostage = orow + s * TT;
#pragma unroll
      for (int tt = 0; tt < TT; tt += 4) {
        float4 ov;
        ov.x = dm_step(rowp[tt + 0], th, nth, recon);
        ov.y = dm_step(rowp[tt + 1], th, nth, recon);
        ov.z = dm_step(rowp[tt + 2], th, nth, recon);
        ov.w = dm_step(rowp[tt + 3], th, nth, recon);
        *(float4*)(ostage + tt) = ov;  // 16B store; lane writes 256B contiguous per stage
      }
    }

    __syncthreads();  // all waves done reading `a` before it is refilled next iter

    const uint32_t t = a; a = b; b = c; c = t;  // rotate ring
  }
}

extern "C" void kernel_launch(void* const* d_in, const int* in_sizes, int n_in,
                              void* d_out, int out_size, void* d_ws, size_t ws_size,
                              hipStream_t stream) {
  (void)in_sizes; (void)n_in; (void)out_size; (void)d_ws; (void)ws_size;
  const float* x   = (const float*)d_in[0];
  const float* thr = (const float*)d_in[1];
  float*       out = (float*)d_out;

  delta_mod_encoder_kernel<<<dim3(NBLK), dim3(TPB), 0, stream>>>(x, thr, out);
}